// Roi_37065567764908
// MI455X (gfx1250) — compile-verified
//
#include <hip/hip_runtime.h>

// ROI-align (bilinear crop 42x42 + 2x2 maxpool -> 21x21) for MI455X / gfx1250.
//
// Strategy: per (ROI, 8-channel tile) workgroup, DMA the ROI's feature-map
// bounding box (<=48x48 px, 8 channels) into LDS with one TENSOR_LOAD_TO_LDS
// (Tensor Data Mover, TENSORcnt), then do all 16 bilinear taps per pooled
// output from LDS. ~74 KB LDS/WG -> 4 WGs (32 waves) per WGP for latency
// hiding. Output (144.5 MB) is streamed with non-temporal stores so the
// 16 MB feature map stays resident in the 192 MB L2.

#define BB   4
#define CC   64
#define HH   128
#define WW   128
#define NNR  320
#define SSP  42     // PREPOOL
#define PPL  21     // POOL
#define CT   8      // channels per tile
#define NCT  (CC / CT)   // 8 tiles
#define MAXB 48     // max bounding-box side (scaled ROI span <= 44 px, +4 guard)

typedef __attribute__((ext_vector_type(4))) unsigned int v4u;
typedef __attribute__((ext_vector_type(4))) int          v4i;
typedef __attribute__((ext_vector_type(8))) int          v8i;

__device__ __forceinline__ unsigned rfl(unsigned x) {
  return (unsigned)__builtin_amdgcn_readfirstlane((int)x);
}

__global__ __launch_bounds__(256)
void roi_align_tdm_kernel(const float* __restrict__ fm,
                          const float* __restrict__ props,
                          const int* __restrict__ ohp,
                          const int* __restrict__ owp,
                          float* __restrict__ out)
{
  __shared__ float tile[CT * MAXB * MAXB];   // [c][by][bx], row stride = bw (compact TDM order)
  __shared__ float fwx0[SSP], fwx1[SSP], fwy0[SSP], fwy1[SSP];
  __shared__ int   xi0[SSP], xi1[SSP], yi0[SSP], yi1[SSP];

  const int tid = threadIdx.x;
  int blk = blockIdx.x;
  const int ct = blk & (NCT - 1); blk >>= 3;
  const int n  = blk % NNR;
  const int b  = blk / NNR;

  // --- scaled ROI (uniform across block) ---
  const float scx = (float)WW / (float)owp[0];
  const float scy = (float)HH / (float)ohp[0];
  const float* pr = props + ((size_t)b * NNR + n) * 4;
  const float x1 = pr[0] * scx, y1 = pr[1] * scy;
  const float x2 = pr[2] * scx, y2 = pr[3] * scy;

  // Bounding box of all bilinear corners (pad 1 low / 2 high for fp-floor slop)
  int bx0 = (int)floorf(x1) - 1; bx0 = bx0 < 0 ? 0 : bx0;
  int bx1 = (int)floorf(x2) + 2; bx1 = bx1 > WW - 1 ? WW - 1 : bx1;
  int by0 = (int)floorf(y1) - 1; by0 = by0 < 0 ? 0 : by0;
  int by1 = (int)floorf(y2) + 2; by1 = by1 > HH - 1 ? HH - 1 : by1;
  int bw = bx1 - bx0 + 1; if (bw > MAXB) bw = MAXB; if (bw < 1) bw = 1;
  int bh = by1 - by0 + 1; if (bh > MAXB) bh = MAXB; if (bh < 1) bh = 1;

  // --- Phase A: wave 0 issues one TDM load of the box tile (x=bw, y=bh, z=CT channels) ---
  if (tid < 32) {
    unsigned ldsb = (unsigned)(size_t)(void*)tile;  // LDS byte address = low 32 bits
    unsigned long long ga = (unsigned long long)(size_t)fm
        + 4ull * ((unsigned long long)(b * CC + ct * CT) * (HH * WW)
                  + (unsigned long long)by0 * WW + (unsigned long long)bx0);

    const unsigned ubw = rfl((unsigned)bw);
    const unsigned ubh = rfl((unsigned)bh);

    // D# group 0: count=1 | lds_addr | global_addr | type=2
    v4u g0;
    g0.x = 1u;
    g0.y = rfl(ldsb);
    g0.z = rfl((unsigned)(ga & 0xFFFFFFFFull));
    g0.w = rfl(((unsigned)(ga >> 32) & 0x1FFFFFFu) | (2u << 30));

    // D# group 1: data_size=2 (4B); tensor dims == tile dims (no clipping);
    // stride0 = W = 128 elems (row), stride1 = H*W = 16384 elems (channel)
    v8i g1;
    g1[0] = (int)(2u << 16);                         // data_size=4B, no pad/iter/multicast
    g1[1] = (int)(ubw << 16);                        // tensor_dim0[15:0]
    g1[2] = (int)(ubh << 16);                        // tensor_dim0[31:16]=0 | tensor_dim1[15:0]
    g1[3] = (int)(ubw << 16);                        // tensor_dim1[31:16]=0 | tile_dim0
    g1[4] = (int)(ubh | ((unsigned)CT << 16));       // tile_dim1 | tile_dim2
    g1[5] = WW;                                      // tensor_dim0_stride[31:0]
    g1[6] = (int)(((unsigned)(HH * WW) & 0xFFFFu) << 16); // str0[47:32]=0 | str1[15:0]
    g1[7] = (int)((unsigned)(HH * WW) >> 16);        // tensor_dim1_stride[47:16]

    v4i g2; g2.x = CT; g2.y = 1; g2.z = 0; g2.w = 0; // tensor_dim2=CT, dim3=1, tile_dim3=0
    v4i g3; g3.x = 0;  g3.y = 0; g3.z = 0; g3.w = 0;

#if defined(__clang_major__) && (__clang_major__ >= 23)
    v8i z8; z8[0]=0; z8[1]=0; z8[2]=0; z8[3]=0; z8[4]=0; z8[5]=0; z8[6]=0; z8[7]=0;
    __builtin_amdgcn_tensor_load_to_lds(g0, g1, g2, g3, z8, 0);
#else
    __builtin_amdgcn_tensor_load_to_lds(g0, g1, g2, g3, 0);
#endif
  }

  // --- Phase B: sample-grid weight tables (match reference arithmetic) ---
  if (tid < SSP) {
    const float base = -1.0f + 2.0f * (float)tid / (float)(SSP - 1);
    // x axis
    {
      float gx = ((x2 - x1) / (float)(WW - 1)) * base
               + ((x1 + x2 - (float)(WW - 1)) / (float)(WW - 1));
      float ix = (gx + 1.0f) * 0.5f * (float)(WW - 1);
      float xf = floorf(ix);
      int   X0 = (int)xf;
      float w1 = ix - xf, w0 = 1.0f - w1;
      fwx0[tid] = (X0 >= 0 && X0 < WW)         ? w0 : 0.0f;
      fwx1[tid] = (X0 + 1 >= 0 && X0 + 1 < WW) ? w1 : 0.0f;
      int r0 = X0 - bx0;     r0 = r0 < 0 ? 0 : (r0 > bw - 1 ? bw - 1 : r0);
      int r1 = X0 + 1 - bx0; r1 = r1 < 0 ? 0 : (r1 > bw - 1 ? bw - 1 : r1);
      xi0[tid] = r0; xi1[tid] = r1;
    }
    // y axis
    {
      float gy = ((y2 - y1) / (float)(HH - 1)) * base
               + ((y1 + y2 - (float)(HH - 1)) / (float)(HH - 1));
      float iy = (gy + 1.0f) * 0.5f * (float)(HH - 1);
      float yf = floorf(iy);
      int   Y0 = (int)yf;
      float w1 = iy - yf, w0 = 1.0f - w1;
      fwy0[tid] = (Y0 >= 0 && Y0 < HH)         ? w0 : 0.0f;
      fwy1[tid] = (Y0 + 1 >= 0 && Y0 + 1 < HH) ? w1 : 0.0f;
      int r0 = Y0 - by0;     r0 = r0 < 0 ? 0 : (r0 > bh - 1 ? bh - 1 : r0);
      int r1 = Y0 + 1 - by0; r1 = r1 < 0 ? 0 : (r1 > bh - 1 ? bh - 1 : r1);
      yi0[tid] = r0; yi1[tid] = r1;
    }
  }

  if (tid < 32) __builtin_amdgcn_s_wait_tensorcnt(0);
  __syncthreads();

  // --- Phase C: bilinear taps from LDS + 2x2 maxpool, NT coalesced stores ---
  const int chS = bw * bh;
  const size_t obase = (((size_t)b * NNR + n) * CC + (size_t)ct * CT) * (PPL * PPL);

  for (int e = tid; e < CT * PPL * PPL; e += 256) {
    const int c  = e / (PPL * PPL);
    const int r  = e - c * (PPL * PPL);
    const int py = r / PPL;
    const int px = r - py * PPL;
    const float* tc = tile + c * chS;

    float m = -3.402823466e+38f;
#pragma unroll
    for (int dy = 0; dy < 2; ++dy) {
      const int  syi = 2 * py + dy;
      const int  ya  = yi0[syi] * bw, yb = yi1[syi] * bw;
      const float wa = fwy0[syi],     wb = fwy1[syi];
#pragma unroll
      for (int dx = 0; dx < 2; ++dx) {
        const int  sxi = 2 * px + dx;
        const int  xa  = xi0[sxi],  xb = xi1[sxi];
        const float ua = fwx0[sxi], ub = fwx1[sxi];
        const float v = wa * (ua * tc[ya + xa] + ub * tc[ya + xb])
                      + wb * (ua * tc[yb + xa] + ub * tc[yb + xb]);
        m = fmaxf(m, v);
      }
    }
    // Output is write-once / never re-read: non-temporal keeps fm hot in L2.
    __builtin_nontemporal_store(m, &out[obase + (size_t)e]);
  }
}

extern "C" void kernel_launch(void* const* d_in, const int* in_sizes, int n_in,
                              void* d_out, int out_size, void* d_ws, size_t ws_size,
                              hipStream_t stream) {
  (void)in_sizes; (void)n_in; (void)out_size; (void)d_ws; (void)ws_size;
  const float* fm  = (const float*)d_in[0];
  const float* pr  = (const float*)d_in[1];
  const int*   oh  = (const int*)d_in[2];
  const int*   ow  = (const int*)d_in[3];
  float*       out = (float*)d_out;

  const int grid = BB * NNR * NCT;  // 10240 blocks: one ROI x 8-channel tile each
  roi_align_tdm_kernel<<<grid, 256, 0, stream>>>(fm, pr, oh, ow, out);
}